// Decoder_34634616275019
// MI455X (gfx1250) — compile-verified
//
#include <hip/hip_runtime.h>
#include <math.h>

// ---------------------------------------------------------------------------
// Problem constants (from the reference)
// ---------------------------------------------------------------------------
#define H    128
#define HID  512
#define CLIP 10.0f
// 1/sqrt(512)
#define INV_SQRT_DH 0.044194173824159216f

// Workspace layout (float slots)
#define WS_U     0     // u[128]
#define WS_S0    128   // s0
#define WS_GMAX  129   // encoded uint max of attn (softmax stabilizer)
#define WS_Z     130   // float sum of exp(attn - M)
#define WS_BEST  132   // uint64 packed (enc(attn)<<32)|(~n), 8B aligned
#define WS_ATTN  256   // attn[N]

typedef float v2f __attribute__((ext_vector_type(2)));
typedef float v8f __attribute__((ext_vector_type(8)));

// Order-preserving float -> uint encoding (for atomicMax on floats)
__device__ __forceinline__ unsigned enc_f32(float f) {
    unsigned u = __float_as_uint(f);
    return (u & 0x80000000u) ? ~u : (u | 0x80000000u);
}
__device__ __forceinline__ float dec_f32(unsigned u) {
    unsigned b = (u & 0x80000000u) ? (u & 0x7FFFFFFFu) : ~u;
    return __uint_as_float(b);
}

// ---------------------------------------------------------------------------
// Stage 1: phi1 = W1 @ v_i + b1 ; u = W2^T @ phi1 ; s0 = phi1 . b2
// Single block of 512 threads. Also re-initializes reduction scratch.
// ---------------------------------------------------------------------------
__global__ void __launch_bounds__(HID)
stage1_kernel(const float* __restrict__ output,
              const float* __restrict__ W1, const float* __restrict__ b1,
              const float* __restrict__ W2, const float* __restrict__ b2,
              const int*   __restrict__ prev_node,
              float* __restrict__ ws) {
    __shared__ float vi[H];
    __shared__ float phi1[HID];
    __shared__ float red[HID];

    const int t = threadIdx.x;
    const int pn = *prev_node;
    if (t < H) vi[t] = output[(size_t)pn * H + t];
    __syncthreads();

    // phi1[t] = W1[t,:] . v_i + b1[t]
    float acc = b1[t];
    #pragma unroll 8
    for (int k = 0; k < H; ++k) acc += W1[(size_t)t * H + k] * vi[k];
    phi1[t] = acc;

    // s0 = sum_h phi1[h] * b2[h]
    red[t] = acc * b2[t];
    __syncthreads();
    for (int s = HID / 2; s > 0; s >>= 1) {
        if (t < s) red[t] += red[t + s];
        __syncthreads();
    }

    // u[k] = sum_h phi1[h] * W2[h,k]   (coalesced across k)
    if (t < H) {
        float s = 0.0f;
        #pragma unroll 8
        for (int h = 0; h < HID; ++h) s += phi1[h] * W2[(size_t)h * H + t];
        ws[WS_U + t] = s;
    }
    if (t == 0) {
        ws[WS_S0] = red[0];
        ((unsigned*)ws)[WS_GMAX] = 0u;               // min of encoding
        ws[WS_Z] = 0.0f;
        *((unsigned long long*)(ws + WS_BEST)) = 0ull;
    }
}

// ---------------------------------------------------------------------------
// Stage 2: per-wave WMMA batch of 16 node dot products, attn + global max.
// Wave w handles nodes [16w, 16w+16). Block = 256 threads = 8 waves.
//
// D = A x B + C with V_WMMA_F32_16X16X4_F32, accumulated over 32 K-chunks.
//   A[i][k] = output[node0+i][4*kc + k]     (per-lane: row = lane&15,
//                                            k = 2*(lane>>4) + {0,1})
//   B[k][j] = u[4*kc + k]  (broadcast over j; same per-lane K split as A)
// => D[i][j] = dot(output[node0+i], u) for every j; read column j=0 from
//    lanes 0 (M=0..7 in VGPR0..7) and 16 (M=8..15 in VGPR0..7).
// ---------------------------------------------------------------------------
__global__ void __launch_bounds__(256)
stage2_kernel(const float* __restrict__ output,
              const float* __restrict__ adj,
              float* __restrict__ ws, int N) {
    __shared__ float su[H];
    const int t = threadIdx.x;
    if (t < H) su[t] = ws[WS_U + t];
    __syncthreads();

    const float s0 = ws[WS_S0];
    float* __restrict__ attn = ws + WS_ATTN;
    unsigned* gmax = ((unsigned*)ws) + WS_GMAX;

    const int lane  = t & 31;
    const int wave  = blockIdx.x * (blockDim.x >> 5) + (t >> 5);
    const int node0 = wave * 16;
    if (node0 >= N) return;            // wave-uniform exit (EXEC stays all-1s)

    const int hi  = lane >> 4;         // 0: K={0,1}+4kc, 1: K={2,3}+4kc
    int row = node0 + (lane & 15);
    if (row >= N) row = N - 1;         // tail: duplicate last row (stores guarded)

    const float* rp = output + (size_t)row * H + 2 * hi;
    const float* up = su + 2 * hi;

    v8f acc = {};
    #pragma unroll 8
    for (int kc = 0; kc < 32; ++kc) {
        v2f a = *(const v2f*)(rp + kc * 4);
        v2f b = *(const v2f*)(up + kc * 4);
        acc = __builtin_amdgcn_wmma_f32_16x16x4_f32(
            /*neg_a=*/false, a, /*neg_b=*/false, b,
            /*c_mod=*/(short)0, acc, /*reuse_a=*/false, /*reuse_b=*/false);
    }

    // Column 0 of D: lane 0 -> rows node0..node0+7, lane 16 -> node0+8..+15
    if ((lane & 15) == 0) {
        const int mbase = node0 + 8 * hi;
        float lmax = -__builtin_inff();
        #pragma unroll
        for (int i = 0; i < 8; ++i) {
            const int n = mbase + i;
            if (n < N) {
                const float av = adj[n];
                const float sc = (s0 + av * acc[i]) * INV_SQRT_DH;
                const float at = CLIP * tanhf(sc) * av;
                attn[n] = at;
                lmax = fmaxf(lmax, at);
            }
        }
        atomicMax(gmax, enc_f32(lmax));
    }
}

// ---------------------------------------------------------------------------
// Pass 3: Z = sum exp(attn - M); masked argmax packed for first-index ties.
// ---------------------------------------------------------------------------
__global__ void __launch_bounds__(256)
pass3_kernel(float* __restrict__ ws, int N) {
    __shared__ float ssum[256];
    __shared__ unsigned long long smax[256];

    const float* attn = ws + WS_ATTN;
    const float M = dec_f32(((const unsigned*)ws)[WS_GMAX]);

    const int t = threadIdx.x;
    const int n = blockIdx.x * blockDim.x + t;

    float e = 0.0f;
    unsigned long long pk = 0ull;
    if (n < N) {
        const float a = attn[n];
        e = expf(a - M);
        if (a != 0.0f) {
            pk = ((unsigned long long)enc_f32(a) << 32) |
                 (unsigned long long)(0xFFFFFFFFu - (unsigned)n);
        }
    }
    ssum[t] = e;
    smax[t] = pk;
    __syncthreads();
    for (int s = 128; s > 0; s >>= 1) {
        if (t < s) {
            ssum[t] += ssum[t + s];
            smax[t] = (smax[t + s] > smax[t]) ? smax[t + s] : smax[t];
        }
        __syncthreads();
    }
    if (t == 0) {
        atomicAdd(ws + WS_Z, ssum[0]);
        atomicMax((unsigned long long*)(ws + WS_BEST), smax[0]);
    }
}

// ---------------------------------------------------------------------------
// Finalize: p = exp(best - M) / Z ; out = (selected_node, p)
// ---------------------------------------------------------------------------
__global__ void finalize_kernel(const float* __restrict__ ws,
                                float* __restrict__ out) {
    if (threadIdx.x == 0 && blockIdx.x == 0) {
        const unsigned long long b = *((const unsigned long long*)(ws + WS_BEST));
        if (b == 0ull) {                  // all attn == 0 -> w all zero
            out[0] = 0.0f;
            out[1] = 0.0f;
        } else {
            const unsigned n = 0xFFFFFFFFu - (unsigned)(b & 0xFFFFFFFFull);
            const float a = dec_f32((unsigned)(b >> 32));
            const float M = dec_f32(((const unsigned*)ws)[WS_GMAX]);
            const float Z = ws[WS_Z];
            out[0] = (float)n;
            out[1] = expf(a - M) / Z;
        }
    }
}

// ---------------------------------------------------------------------------
// Launcher
// ---------------------------------------------------------------------------
extern "C" void kernel_launch(void* const* d_in, const int* in_sizes, int n_in,
                              void* d_out, int out_size, void* d_ws, size_t ws_size,
                              hipStream_t stream) {
    const float* output = (const float*)d_in[0];
    const float* adj    = (const float*)d_in[1];
    const float* W1     = (const float*)d_in[2];
    const float* b1     = (const float*)d_in[3];
    const float* W2     = (const float*)d_in[4];
    const float* b2     = (const float*)d_in[5];
    const int*   prev   = (const int*)  d_in[6];

    float* ws  = (float*)d_ws;
    float* out = (float*)d_out;
    const int N = in_sizes[1];          // adj has N elements

    stage1_kernel<<<1, HID, 0, stream>>>(output, W1, b1, W2, b2, prev, ws);

    const int nwaves  = (N + 15) / 16;
    const int nblocks = (nwaves + 7) / 8;          // 8 waves per 256-thread block
    stage2_kernel<<<nblocks, 256, 0, stream>>>(output, adj, ws, N);

    pass3_kernel<<<(N + 255) / 256, 256, 0, stream>>>(ws, N);

    finalize_kernel<<<1, 32, 0, stream>>>(ws, out);
}